// ConceptCodebook_81277961109953
// MI455X (gfx1250) — compile-verified
//
#include <hip/hip_runtime.h>
#include <hip/hip_bf16.h>
#include <math.h>

// ---------------------------------------------------------------------------
// VQ codebook lookup for MI455X (gfx1250).
//   z:   [N=32768, D=768] f32,  emb: [K=2048, D=768] f32
//   out: z_q [N*D] f32, then vq_loss (1) and perplexity (1).
// Compute-bound f32 GEMM (103 GFLOP vs ~206MB traffic) -> f32 WMMA path,
// async global->LDS staging, fused argmin in accumulator registers.
// ---------------------------------------------------------------------------

#define DDIM   768
#define KCODES 2048

typedef float v2f __attribute__((ext_vector_type(2)));
typedef float v8f __attribute__((ext_vector_type(8)));

// --------------------------------------------------------------------------
// Kernel 1: eNorms[k] = sum_d e[k][d]^2  (one wave32 per code row) + zero hist
// --------------------------------------------------------------------------
__global__ __launch_bounds__(256) void enorm_init_kernel(
    const float* __restrict__ emb, float* __restrict__ eNorms,
    int* __restrict__ hist)
{
    const int tid  = threadIdx.x;
    const int lane = tid & 31;
    const int wave = tid >> 5;
    const int row  = blockIdx.x * 8 + wave;

    const float* e = emb + (size_t)row * DDIM;
    float s = 0.0f;
    #pragma unroll
    for (int d = lane; d < DDIM; d += 32) { float x = e[d]; s += x * x; }
    #pragma unroll
    for (int off = 16; off >= 1; off >>= 1) s += __shfl_xor(s, off, 32);
    if (lane == 0) eNorms[row] = s;

    const int g = blockIdx.x * 256 + tid;
    if (g < KCODES) hist[g] = 0;
}

// --------------------------------------------------------------------------
// Kernel 2: fused GEMM + argmin via v_wmma_f32_16x16x4_f32.
// Block (8 waves) owns 16 rows x all 2048 codes.
//   - 16x768 z tile staged via GLOBAL_LOAD_ASYNC_TO_LDS_B128 (ASYNCcnt).
//   - each wave sweeps 8 pairs of column tiles; one A fragment feeds 2 WMMAs.
//   - dist = ||e||^2 - 2*acc  (||z||^2 is row-constant: irrelevant to argmin).
//   - per-lane strict-< running min (columns visited in increasing order ->
//     earliest index kept on ties), shfl_xor reduce over 16-lane halves,
//     cross-wave LDS reduce.
// grid = N/16 blocks.
// --------------------------------------------------------------------------
__global__ __launch_bounds__(256) void vq_argmin_kernel(
    const float* __restrict__ z, const float* __restrict__ emb,
    const float* __restrict__ eNorms, int* __restrict__ ids,
    int* __restrict__ hist)
{
    __shared__ __align__(16) float ldsA[16 * DDIM];   // 48 KB z tile
    __shared__ float redV[8][16];
    __shared__ int   redI[8][16];

    const int tid  = threadIdx.x;
    const int wave = tid >> 5;
    const int lane = tid & 31;
    const int n16  = lane & 15;     // A: row m; B/C: column within tile
    const int kh   = lane >> 4;     // k-half selector (ISA 16x4 f32 layout)
    const size_t rowBase = (size_t)blockIdx.x * 16;

    // ---- async-stage A tile: 16 rows x 768 f32 = 3072 x b128, 12/thread ----
    {
        unsigned ldsOff = (unsigned)(size_t)(&ldsA[0]) + (unsigned)tid * 16u;
        unsigned long long gAddr =
            (unsigned long long)(const void*)((const char*)(z + rowBase * DDIM)
                                              + (size_t)tid * 16u);
        // INST_OFFSET is added to BOTH the LDS and global addresses (ISA 10.x)
        asm volatile(
            "global_load_async_to_lds_b128 %0, %1, off\n\t"
            "global_load_async_to_lds_b128 %0, %1, off offset:4096\n\t"
            "global_load_async_to_lds_b128 %0, %1, off offset:8192\n\t"
            "global_load_async_to_lds_b128 %0, %1, off offset:12288\n\t"
            "global_load_async_to_lds_b128 %0, %1, off offset:16384\n\t"
            "global_load_async_to_lds_b128 %0, %1, off offset:20480\n\t"
            "global_load_async_to_lds_b128 %0, %1, off offset:24576\n\t"
            "global_load_async_to_lds_b128 %0, %1, off offset:28672\n\t"
            "global_load_async_to_lds_b128 %0, %1, off offset:32768\n\t"
            "global_load_async_to_lds_b128 %0, %1, off offset:36864\n\t"
            "global_load_async_to_lds_b128 %0, %1, off offset:40960\n\t"
            "global_load_async_to_lds_b128 %0, %1, off offset:45056\n\t"
            "s_wait_asynccnt 0x0"
            :: "v"(ldsOff), "v"(gAddr) : "memory");
    }
    __syncthreads();

    // A fragment: lane (m=n16, k=d+2*kh .. +1) -> contiguous float2 in LDS
    const v2f* aBase = (const v2f*)(ldsA + n16 * DDIM + 2 * kh);

    float best[8];
    int   bidx[8];
    #pragma unroll
    for (int i = 0; i < 8; ++i) { best[i] = 3.4e38f; bidx[i] = 0; }

    for (int t = 0; t < 8; ++t) {
        const int kt0  = wave * 2 + t * 16;     // pair of column tiles
        const int col0 = kt0 * 16 + n16;
        const int col1 = col0 + 16;
        const v2f* bBase0 = (const v2f*)(emb + (size_t)col0 * DDIM + 2 * kh);
        const v2f* bBase1 = (const v2f*)(emb + (size_t)col1 * DDIM + 2 * kh);

        if (t < 7) {  // prefetch next pair's embedding rows (L2-resident)
            __builtin_prefetch(emb + (size_t)(col0 + 256) * DDIM, 0, 0);
            __builtin_prefetch(emb + (size_t)(col1 + 256) * DDIM, 0, 0);
        }

        v8f c0 = {};
        v8f c1 = {};
        #pragma unroll 4
        for (int d = 0; d < DDIM; d += 4) {
            v2f a  = aBase[d >> 1];       // one A fragment feeds two WMMAs
            v2f b0 = bBase0[d >> 1];
            v2f b1 = bBase1[d >> 1];
            c0 = __builtin_amdgcn_wmma_f32_16x16x4_f32(
                     false, a, false, b0, (short)0, c0, false, false);
            c1 = __builtin_amdgcn_wmma_f32_16x16x4_f32(
                     false, a, false, b1, (short)0, c1, false, false);
        }

        const float en0 = eNorms[col0];
        const float en1 = eNorms[col1];
        #pragma unroll
        for (int i = 0; i < 8; ++i) {     // row M = 8*kh + i
            const float d0 = en0 - 2.0f * c0[i];
            if (d0 < best[i]) { best[i] = d0; bidx[i] = col0; }
            const float d1 = en1 - 2.0f * c1[i];
            if (d1 < best[i]) { best[i] = d1; bidx[i] = col1; }
        }
    }

    // ---- intra-wave: reduce over 16 lanes holding the same row ----
    #pragma unroll
    for (int i = 0; i < 8; ++i) {
        float v = best[i]; int id = bidx[i];
        #pragma unroll
        for (int off = 8; off >= 1; off >>= 1) {
            const float ov = __shfl_xor(v, off, 16);
            const int   oi = __shfl_xor(id, off, 16);
            if (ov < v || (ov == v && oi < id)) { v = ov; id = oi; }
        }
        if (n16 == 0) { redV[wave][kh * 8 + i] = v; redI[wave][kh * 8 + i] = id; }
    }
    __syncthreads();

    // ---- cross-wave: 8 candidates per row -> final argmin ----
    if (tid < 16) {
        float v = redV[0][tid]; int id = redI[0][tid];
        #pragma unroll
        for (int w = 1; w < 8; ++w) {
            const float ov = redV[w][tid]; const int oi = redI[w][tid];
            if (ov < v || (ov == v && oi < id)) { v = ov; id = oi; }
        }
        ids[rowBase + tid] = id;
        atomicAdd(&hist[id], 1);
    }
}

// --------------------------------------------------------------------------
// Kernel 3: z_q[n,:] = emb[ids[n],:]  (one block per token, float4 streams)
// --------------------------------------------------------------------------
__global__ __launch_bounds__(192) void gather_kernel(
    const float* __restrict__ emb, const int* __restrict__ ids,
    float* __restrict__ zq)
{
    const int n  = blockIdx.x;
    const int id = ids[n];
    const float4* src = (const float4*)(emb + (size_t)id * DDIM);
    float4*       dst = (float4*)(zq + (size_t)n * DDIM);
    dst[threadIdx.x] = src[threadIdx.x];
}

// --------------------------------------------------------------------------
// Kernel 4: perplexity = exp(-sum p*log(p+1e-10)), vq_loss = 0  (one block)
// --------------------------------------------------------------------------
__global__ __launch_bounds__(256) void perplexity_kernel(
    const int* __restrict__ hist, float* __restrict__ out_tail, int n_tokens)
{
    __shared__ float red[256];
    const float invN = 1.0f / (float)n_tokens;
    float s = 0.0f;
    for (int k = threadIdx.x; k < KCODES; k += 256) {
        const float p = (float)hist[k] * invN;
        s += p * logf(p + 1e-10f);
    }
    red[threadIdx.x] = s;
    __syncthreads();
    #pragma unroll
    for (int st = 128; st >= 1; st >>= 1) {
        if (threadIdx.x < st) red[threadIdx.x] += red[threadIdx.x + st];
        __syncthreads();
    }
    if (threadIdx.x == 0) {
        out_tail[0] = 0.0f;           // vq_loss (eval mode)
        out_tail[1] = expf(-red[0]);  // perplexity
    }
}

// --------------------------------------------------------------------------
extern "C" void kernel_launch(void* const* d_in, const int* in_sizes, int n_in,
                              void* d_out, int out_size, void* d_ws, size_t ws_size,
                              hipStream_t stream)
{
    const float* z   = (const float*)d_in[0];   // [N, 768]
    const float* emb = (const float*)d_in[1];   // [2048, 768]
    float* out = (float*)d_out;

    const int N = in_sizes[0] / DDIM;           // 32768 tokens

    // workspace: eNorms[2048] f32 | hist[2048] i32 | ids[N] i32  (~144 KB)
    float* eNorms = (float*)d_ws;
    int*   hist   = (int*)((char*)d_ws + KCODES * sizeof(float));
    int*   ids    = (int*)((char*)d_ws + KCODES * sizeof(float)
                                       + KCODES * sizeof(int));

    enorm_init_kernel<<<KCODES / 8, 256, 0, stream>>>(emb, eNorms, hist);
    vq_argmin_kernel<<<N / 16, 256, 0, stream>>>(z, emb, eNorms, ids, hist);
    gather_kernel<<<N, 192, 0, stream>>>(emb, ids, out);
    perplexity_kernel<<<1, 256, 0, stream>>>(hist, out + (size_t)N * DDIM, N);
}